// Adder2D_83683142795456
// MI455X (gfx1250) — compile-verified
//
#include <hip/hip_runtime.h>
#include <stdint.h>

// Problem constants (reference: x (8,128,16,16) f32, W (128,128,3,3) f32,
// stride=1, pad=1 -> out (8,128,16,16) f32).
#define N_      8
#define C_      128
#define F_      128
#define D_      1152        // C_ * 3 * 3
#define FT      8           // filters per block (per-thread accumulators)
#define CCH     16          // channels per LDS chunk
#define NCHUNK  (C_ / CCH)  // 8

// CDNA5 async global -> LDS copy (ASYNCcnt-tracked). LDS dest is a 32-bit
// byte offset in a VGPR, global source is a 64-bit address in a VGPR pair.
__device__ __forceinline__ void async_ld_b128(uint32_t lds_off, const float* gsrc) {
  asm volatile("global_load_async_to_lds_b128 %0, %1, off"
               :: "v"(lds_off), "v"(gsrc)
               : "memory");
}

__device__ __forceinline__ void wait_async0() {
#if __has_builtin(__builtin_amdgcn_s_wait_asynccnt)
  __builtin_amdgcn_s_wait_asynccnt(0);
#else
  asm volatile("s_wait_asynccnt 0" ::: "memory");
#endif
}

__launch_bounds__(256)
__global__ void adder2d_l1(const float* __restrict__ x,
                           const float* __restrict__ Wt,
                           float* __restrict__ out) {
  // Double-buffered: 2 x (16 KB x-chunk + 6 KB padded W-chunk) = 44 KB
  // of the 320 KB WGP LDS.
  __shared__ __align__(16) float xsf[2][CCH * 256];      // [buf][cl][h*16+w]
  __shared__ __align__(16) float wsf[2][CCH * 12 * FT];  // [buf][cl][tap(12,9 used)][f]

  const int n   = blockIdx.x;          // batch
  const int f0  = blockIdx.y * FT;     // filter tile base
  const int tid = threadIdx.x;         // 256 threads = one 16x16 output plane
  const int ho  = tid >> 4;
  const int wo  = tid & 15;

  // Per-thread tap offsets + zero-padding mask (pad taps contribute |W|).
  int  xoff[9];
  bool xin[9];
#pragma unroll
  for (int r = 0; r < 9; ++r) {
    const int hi = ho + (r / 3) - 1;
    const int wi = wo + (r % 3) - 1;
    const bool inb = ((unsigned)hi < 16u) && ((unsigned)wi < 16u);
    xin[r]  = inb;
    xoff[r] = inb ? (hi * 16 + wi) : 0;
  }

  float acc[FT];
#pragma unroll
  for (int f = 0; f < FT; ++f) acc[f] = 0.0f;

  // Low 32 bits of a generic LDS address == wave-relative LDS byte offset.
  const uint32_t xs_base0 = (uint32_t)(uintptr_t)(&xsf[0][0]);
  const uint32_t xs_base1 = (uint32_t)(uintptr_t)(&xsf[1][0]);

  // ---- stage(chunk cc -> buffer b): async x tile + f-interleaved W tile ----
  auto stage = [&](int cc, int b) {
    // x chunk: 16 ch * 256 f32 = 1024 x 16 B async transfers (4 per thread)
    const float*   xsrc    = x + ((size_t)n * C_ + (size_t)cc * CCH) * 256;
    const uint32_t xs_base = b ? xs_base1 : xs_base0;
#pragma unroll
    for (int t = 0; t < 4; ++t) {
      const int i = tid + t * 256;                 // transfer id 0..1023
      async_ld_b128(xs_base + (uint32_t)i * 16u, xsrc + (size_t)i * 4);
    }
    // W chunk into [cl][12][FT]: 8-filter groups 32B-aligned for ds_load_b128
    for (int i = tid; i < FT * CCH * 9; i += 256) {  // 1152 elements
      const int f  = i & (FT - 1);
      const int q  = i >> 3;          // cl*9 + r  (chunk-local d index)
      const int cl = q / 9;
      const int r  = q - cl * 9;
      wsf[b][(cl * 12 + r) * FT + f] =
          Wt[(size_t)(f0 + f) * D_ + (size_t)cc * (CCH * 9) + q];
    }
  };

  // Prologue: fill buffer 0 with chunk 0.
  stage(0, 0);
  wait_async0();
  __syncthreads();

  for (int cc = 0; cc < NCHUNK; ++cc) {
    const int cur = cc & 1;

    // Prefetch next chunk into the other buffer; the async engine runs
    // while the VALU loop below chews through the current buffer.
    if (cc + 1 < NCHUNK) stage(cc + 1, cur ^ 1);

    // ---- VALU core: acc[f] += |W[f,d] - x[d]| over this chunk ----
    const float* xbuf = &xsf[cur][0];
    const float* wbuf = &wsf[cur][0];
    for (int cl = 0; cl < CCH; ++cl) {
      const float* xp = xbuf + cl * 256;
      const float* wr = wbuf + cl * 12 * FT;
#pragma unroll
      for (int r = 0; r < 9; ++r) {
        const float xv = xin[r] ? xp[xoff[r]] : 0.0f;   // 1 ds_load + cndmask
        const float4 w0 = *reinterpret_cast<const float4*>(wr + r * FT);      // ds_load_b128
        const float4 w1 = *reinterpret_cast<const float4*>(wr + r * FT + 4);  // ds_load_b128
        acc[0] += fabsf(w0.x - xv);
        acc[1] += fabsf(w0.y - xv);
        acc[2] += fabsf(w0.z - xv);
        acc[3] += fabsf(w0.w - xv);
        acc[4] += fabsf(w1.x - xv);
        acc[5] += fabsf(w1.y - xv);
        acc[6] += fabsf(w1.z - xv);
        acc[7] += fabsf(w1.w - xv);
      }
    }

    // Drain this wave's prefetch, then block-wide barrier: makes the
    // freshly filled buffer visible and proves the buffer we will refill
    // next iteration has been fully consumed by all waves.
    wait_async0();
    __syncthreads();
  }

  // out[n, f0+f, ho, wo] = -acc[f]
#pragma unroll
  for (int f = 0; f < FT; ++f) {
    out[(((size_t)n * F_ + (f0 + f)) * 16 + ho) * 16 + wo] = -acc[f];
  }
}

extern "C" void kernel_launch(void* const* d_in, const int* in_sizes, int n_in,
                              void* d_out, int out_size, void* d_ws, size_t ws_size,
                              hipStream_t stream) {
  (void)in_sizes; (void)n_in; (void)out_size; (void)d_ws; (void)ws_size;
  const float* x  = (const float*)d_in[0];   // (8,128,16,16) f32
  const float* Wt = (const float*)d_in[1];   // (128,128,3,3) f32
  float* out = (float*)d_out;                // (8,128,16,16) f32

  dim3 grid(N_, F_ / FT);   // 8 x 16 = 128 blocks
  dim3 block(256);          // 8 wave32 per block
  adder2d_l1<<<grid, block, 0, stream>>>(x, Wt, out);
}